// GNNmodel_40965398069501
// MI455X (gfx1250) — compile-verified
//
#include <hip/hip_runtime.h>
#include <hip/hip_bf16.h>

typedef __attribute__((ext_vector_type(2))) float v2f;
typedef __attribute__((ext_vector_type(8))) float v8f;
typedef unsigned int v4u __attribute__((ext_vector_type(4)));
typedef int v8i __attribute__((ext_vector_type(8)));
typedef int v4i __attribute__((ext_vector_type(4)));

#define DIM 128
#define LPITCH 132   // 16x128 LDS tile pitch (pad 4 dwords -> conflict-free column reads)
// dynamic-LDS layout (byte offsets known at compile time; needed for TDM lds_addr)
#define LA1_OFF 0
#define LA2_OFF (16 * LPITCH * 4)           // 8448 B
#define RED_OFF (32 * LPITCH * 4)           // 16896 B
#define SMEM_BYTES (RED_OFF + 16 * 36 * 4)  // 19200 B

// ---------------------------------------------------------------- zero
__global__ void __launch_bounds__(256) zero_kernel(float* __restrict__ p, long n) {
    long i = (long)blockIdx.x * blockDim.x + threadIdx.x;
    long stride = (long)gridDim.x * blockDim.x;
    for (; i < n; i += stride) p[i] = 0.0f;
}

// ------------------------------------------------- edge gather/scatter-add
// one wave32 per edge; lane l covers floats [4l, 4l+3] of the 128-wide row.
__global__ void __launch_bounds__(256) scatter_kernel(
    const float* __restrict__ h, const int* __restrict__ src,
    const int* __restrict__ dst, const float* __restrict__ ew,
    float* __restrict__ agg, int E)
{
    long gid = (long)blockIdx.x * blockDim.x + threadIdx.x;
    int gwave = (int)(gid >> 5);
    int lane  = threadIdx.x & 31;
    if (gwave >= E) return;
    int   s = src[gwave];
    int   d = dst[gwave];
    float w = ew[gwave];
    const float4* row = (const float4*)(h + (size_t)s * DIM);
    float4 v = row[lane];                       // coalesced 512B row read (L2-hot)
    float* out = agg + (size_t)d * DIM + lane * 4;
    unsafeAtomicAdd(out + 0, v.x * w);          // global_atomic_add_f32
    unsafeAtomicAdd(out + 1, v.y * w);
    unsafeAtomicAdd(out + 2, v.z * w);
    unsafeAtomicAdd(out + 3, v.w * w);
}

// Issue one TDM 2D tile load: global [rows x 128 f32, stride 128] -> LDS at
// lds_off with 4-dword padding every 128 dwords (matches LPITCH=132).
__device__ __forceinline__ void tdm_tile_load(const float* gsrc, unsigned lds_off,
                                              unsigned rows) {
    unsigned long long ga = (unsigned long long)(uintptr_t)gsrc;
    v4u g0 = { 1u,                                   // count=1, user descriptor
               lds_off,                              // D#.lds_addr
               (unsigned)ga,                         // global_addr[31:0]
               (unsigned)((ga >> 32) & 0x01FFFFFFu) | (2u << 30) };  // addr[56:32] | type=2
    v8i g1 = { (int)((2u << 16) | (1u << 20) | (6u << 22) | (3u << 25)),
               //  data_size=4B | pad_enable | pad_interval=128dw | pad_amount=4dw
               (int)(128u << 16),                    // tensor_dim0 = 128
               (int)(rows << 16),                    // tensor_dim1 = rows (OOB rows -> 0)
               (int)(128u << 16),                    // tile_dim0 = 128
               16,                                   // tile_dim1 = 16, tile_dim2 = 0
               128,                                  // tensor_dim0_stride = 128
               0, 0 };                               // tensor_dim1_stride unused (2D)
    v4i zz4 = { 0, 0, 0, 0 };
    v8i zz8 = { 0, 0, 0, 0, 0, 0, 0, 0 };
    __builtin_amdgcn_tensor_load_to_lds(g0, g1, zz4, zz4, zz8, 0);
}

// --------------------------------------------------- fused dual-GEMM + LN + act
// block = 16 rows x 128 cols. 8 waves; wave w owns 16x16 tile at cols [16w,16w+15].
// Out = act/LN( A1 @ W1 (+ A2 @ W2) + bias )
// mode 0: LayerNorm then PReLU(alpha)   (graph layers)
// mode 1: ReLU then LayerNorm           (classifier hidden)
__global__ void __launch_bounds__(256) gemm_ln_act(
    const float* __restrict__ A1, const float* __restrict__ W1,
    const float* __restrict__ A2, const float* __restrict__ W2,
    const float* __restrict__ bias, const float* __restrict__ g,
    const float* __restrict__ bvec, const float* __restrict__ alpha_p,
    int mode, float* __restrict__ Out, int N)
{
    extern __shared__ float smem[];
    float* lA1 = smem;                    // [16][LPITCH] at byte 0
    float* lA2 = smem + 16 * LPITCH;      // [16][LPITCH] at byte 8448
    float* red = smem + 32 * LPITCH;      // [16][36]     at byte 16896

    const int t = threadIdx.x;
    const int rowBase = blockIdx.x * 16;

    // --- TDM stage of the A tiles (wave 0 issues the DMA; EXEC ignored by TDM)
    if (t < 32) {
        unsigned rows = (unsigned)((N - rowBase) < 16 ? (N - rowBase) : 16);
        tdm_tile_load(A1 + (size_t)rowBase * DIM, LA1_OFF, rows);
        if (A2) tdm_tile_load(A2 + (size_t)rowBase * DIM, LA2_OFF, rows);
        __builtin_amdgcn_s_wait_tensorcnt(0);
    }
    __syncthreads();

    const int wave  = t >> 5;
    const int lane  = t & 31;
    const int mrow  = lane & 15;          // A row within tile
    const int khalf = (lane >> 4) << 1;   // K sub-offset: 0 or 2
    const int ncol  = (wave << 4) + (lane & 15);  // output column 0..127

    v8f c = {};
    if (A2) {
        for (int k = 0; k < DIM; k += 4) {
            v2f a1, b1, a2, b2;
            a1.x = lA1[mrow * LPITCH + k + khalf];
            a1.y = lA1[mrow * LPITCH + k + khalf + 1];
            const float* wp1 = W1 + (size_t)(k + khalf) * DIM + ncol;
            b1.x = wp1[0]; b1.y = wp1[DIM];
            c = __builtin_amdgcn_wmma_f32_16x16x4_f32(false, a1, false, b1,
                                                      (short)0, c, false, false);
            a2.x = lA2[mrow * LPITCH + k + khalf];
            a2.y = lA2[mrow * LPITCH + k + khalf + 1];
            const float* wp2 = W2 + (size_t)(k + khalf) * DIM + ncol;
            b2.x = wp2[0]; b2.y = wp2[DIM];
            c = __builtin_amdgcn_wmma_f32_16x16x4_f32(false, a2, false, b2,
                                                      (short)0, c, false, false);
        }
    } else {
        for (int k = 0; k < DIM; k += 4) {
            v2f a1, b1;
            a1.x = lA1[mrow * LPITCH + k + khalf];
            a1.y = lA1[mrow * LPITCH + k + khalf + 1];
            const float* wp1 = W1 + (size_t)(k + khalf) * DIM + ncol;
            b1.x = wp1[0]; b1.y = wp1[DIM];
            c = __builtin_amdgcn_wmma_f32_16x16x4_f32(false, a1, false, b1,
                                                      (short)0, c, false, false);
        }
    }

    const float bb = bias[ncol];
    __syncthreads();                 // done reading lA1/lA2; reuse lA1 as C staging
    #pragma unroll
    for (int r = 0; r < 8; r++) {
        int m = r + ((lane >= 16) ? 8 : 0);   // C layout: VGPR r -> rows r / r+8
        lA1[m * LPITCH + ncol] = c[r] + bb;
    }
    __syncthreads();

    // ---- LayerNorm + activation: 16 threads per row, 8 elements each
    const int row = t >> 4;
    const int seg = t & 15;
    const float aslope = alpha_p ? alpha_p[0] : 0.0f;
    float vals[8];
    float s = 0.0f, sq = 0.0f;
    #pragma unroll
    for (int j = 0; j < 8; j++) {
        float x = lA1[row * LPITCH + seg * 8 + j];
        if (mode == 1) x = x > 0.0f ? x : 0.0f;   // ReLU before LN (classifier)
        vals[j] = x; s += x; sq += x * x;
    }
    red[row * 36 + seg]      = s;
    red[row * 36 + 16 + seg] = sq;
    __syncthreads();
    if (seg == 0) {
        float ts = 0.0f, tq = 0.0f;
        #pragma unroll
        for (int j = 0; j < 16; j++) { ts += red[row * 36 + j]; tq += red[row * 36 + 16 + j]; }
        float mu  = ts * (1.0f / DIM);
        float var = tq * (1.0f / DIM) - mu * mu;
        red[row * 36 + 32] = mu;
        red[row * 36 + 33] = __frsqrt_rn(var + 1e-5f);
    }
    __syncthreads();
    const float mu   = red[row * 36 + 32];
    const float rstd = red[row * 36 + 33];

    const int gr = rowBase + row;
    if (gr < N) {
        float4 o[2];
        #pragma unroll
        for (int j = 0; j < 8; j++) {
            int col = seg * 8 + j;
            float y = (vals[j] - mu) * rstd * g[col] + bvec[col];
            if (mode == 0) y = y >= 0.0f ? y : aslope * y;   // PReLU after LN
            ((float*)o)[j] = y;
        }
        float4* op = (float4*)(Out + (size_t)gr * DIM + seg * 8);
        op[0] = o[0]; op[1] = o[1];
    }
}

// ---------------------------------------------------------- 128 -> 2 projection
__global__ void __launch_bounds__(256) logits_kernel(
    const float* __restrict__ z, const float* __restrict__ W2,
    const float* __restrict__ b2, float* __restrict__ out, int N)
{
    long gid = (long)blockIdx.x * blockDim.x + threadIdx.x;
    int gwave = (int)(gid >> 5);
    int lane  = threadIdx.x & 31;
    if (gwave >= N) return;
    const float* zr = z + (size_t)gwave * DIM;
    float a0 = 0.0f, a1 = 0.0f;
    for (int d = lane; d < DIM; d += 32) {
        float x = zr[d];
        a0 += x * W2[d * 2 + 0];
        a1 += x * W2[d * 2 + 1];
    }
    for (int off = 16; off; off >>= 1) {
        a0 += __shfl_down(a0, off, 32);
        a1 += __shfl_down(a1, off, 32);
    }
    if (lane == 0) {
        out[(size_t)gwave * 2 + 0] = a0 + b2[0];
        out[(size_t)gwave * 2 + 1] = a1 + b2[1];
    }
}

// ------------------------------------------------------------------- driver
extern "C" void kernel_launch(void* const* d_in, const int* in_sizes, int n_in,
                              void* d_out, int out_size, void* d_ws, size_t ws_size,
                              hipStream_t stream) {
    const float* features = (const float*)d_in[0];
    const int*   eidx     = (const int*)  d_in[1];
    const float* ew       = (const float*)d_in[2];
    const float* W_rel    = (const float*)d_in[3];
    const float* W_root   = (const float*)d_in[4];
    const float* bias     = (const float*)d_in[5];
    const float* ln_g     = (const float*)d_in[6];
    const float* ln_b     = (const float*)d_in[7];
    const float* alpha    = (const float*)d_in[8];
    const float* cls_W1   = (const float*)d_in[9];
    const float* cls_b1   = (const float*)d_in[10];
    const float* cls_ln_g = (const float*)d_in[11];
    const float* cls_ln_b = (const float*)d_in[12];
    const float* cls_W2   = (const float*)d_in[13];
    const float* cls_b2   = (const float*)d_in[14];

    const int N = in_sizes[0] / DIM;
    const int E = in_sizes[2];
    const int L = in_sizes[8];
    const int* src = eidx;
    const int* dst = eidx + E;

    size_t per = (size_t)N * DIM;
    float* hA  = (float*)d_ws;
    float* hB  = hA + per;
    float* agg = hB + per;
    float* bufs[2] = { hA, hB };

    const int nb = (N + 15) / 16;         // GEMM blocks (16 rows each)
    const int sb = (E + 7) / 8;           // scatter: 8 waves/block, wave per edge
    const int lb = (N + 7) / 8;           // logits: wave per row

    const float* cur = features;
    for (int i = 0; i < L; i++) {
        zero_kernel<<<2048, 256, 0, stream>>>(agg, (long)per);
        scatter_kernel<<<sb, 256, 0, stream>>>(cur, src, dst, ew, agg, E);
        float* out = bufs[i & 1];
        gemm_ln_act<<<nb, 256, SMEM_BYTES, stream>>>(
            agg, W_rel + (size_t)i * DIM * DIM,
            cur, W_root + (size_t)i * DIM * DIM,
            bias + i * DIM, ln_g + i * DIM, ln_b + i * DIM, alpha + i,
            /*mode=*/0, out, N);
        cur = out;
    }
    float* z = bufs[L & 1];               // ping-pong buffer not aliasing `cur`
    gemm_ln_act<<<nb, 256, SMEM_BYTES, stream>>>(
        cur, cls_W1, nullptr, nullptr,
        cls_b1, cls_ln_g, cls_ln_b, nullptr,
        /*mode=*/1, z, N);
    logits_kernel<<<lb, 256, 0, stream>>>(z, cls_W2, cls_b2, (float*)d_out, N);
}